// A_region_compute_46995532153406
// MI455X (gfx1250) — compile-verified
//
#include <hip/hip_runtime.h>
#include <math.h>

// Problem constants (B=2, N=1024, C=64 from the reference)
#define NB   2
#define NN   1024
#define NC   64
#define ROWS (NB * NN)   // 2048 flattened (b, n) rows

typedef __attribute__((ext_vector_type(2))) float v2f;
typedef __attribute__((ext_vector_type(8))) float v8f;

// -----------------------------------------------------------------------------
// Kernel 1: ut[b][c][n] = sum_k W1[c][k] * x[b][n][k]   (u = x @ W1^T, stored
// transposed so kernel 2's j-loop is lane-coalesced). One wave per 16x16 tile,
// fp32 WMMA 16x16x4, K=64 in 16 steps.
//
// Fragment layouts per CDNA5 ISA 7.12.2:
//   A 16x4 : lane l -> M = l&15, K = {0,1} (l<16) or {2,3} (l>=16)  -> v2f
//   B 4x16 : lane l -> N = l&15, K row = vgpr + 2*(l>=16)           -> v2f
//   D 16x16: lane l, vgpr v -> M = v + 8*(l>=16), N = l&15          -> v8f
// -----------------------------------------------------------------------------
__global__ __launch_bounds__(256) void k1_proj_wmma(const float* __restrict__ x,
                                                    const float* __restrict__ w1,
                                                    float* __restrict__ ut) {
  const int wave   = (blockIdx.x * 256 + threadIdx.x) >> 5; // 0..511
  const int lane   = threadIdx.x & 31;
  const int tile_m = wave >> 2;        // 0..127  (rows of 16)
  const int tile_n = wave & 3;         // 0..3    (output channels of 16)
  const int r0     = tile_m * 16;
  const int n0     = tile_n * 16;
  const int m      = lane & 15;
  const int kb     = (lane >> 4) << 1; // 0 or 2

  const float* xrow = x  + (size_t)(r0 + m) * NC; // A row (this lane's M)
  const float* wrow = w1 + (size_t)(n0 + m) * NC; // W1[cout][k], cout = n0 + (lane&15)

  v8f acc = {};
#pragma unroll
  for (int k0 = 0; k0 < NC; k0 += 4) {
    v2f a = *(const v2f*)(xrow + k0 + kb);
    v2f b = *(const v2f*)(wrow + k0 + kb);
    // (neg_a, A, neg_b, B, c_mod, C, reuse_a, reuse_b)
    acc = __builtin_amdgcn_wmma_f32_16x16x4_f32(false, a, false, b,
                                                (short)0, acc, false, false);
  }

  // Scatter-store D transposed: ut[(b*NC + cout)*NN + n]
  const int cout  = n0 + (lane & 15);
  const int rbase = r0 + ((lane >> 4) << 3); // +8 for upper lane half
#pragma unroll
  for (int v = 0; v < 8; ++v) {
    const int r = rbase + v;          // flattened row = b*NN + n (tiles never cross b)
    const int b = r >> 10;
    const int n = r & (NN - 1);
    ut[((size_t)b * NC + cout) * NN + n] = acc[v];
  }
}

// -----------------------------------------------------------------------------
// Kernel 2: per block, TI=8 consecutive i-rows; scores
//   s[i][j] = sum_c w2[c] * max(u_i[c] + b1[c] - u_j[c], 0)
// then a numerically-stable softmax over j. (b2 dropped: softmax is invariant
// to a constant shift.) 256 threads; each thread owns j = tid + {0,256,512,768}
// for all 8 i-rows -> 32 score registers. ut reads are lane-coalesced.
// -----------------------------------------------------------------------------
#define TI 8

__global__ __launch_bounds__(256) void k2_scores_softmax(const float* __restrict__ ut,
                                                         const float* __restrict__ b1,
                                                         const float* __restrict__ w2,
                                                         float* __restrict__ out) {
  __shared__ float s_vi[TI][NC];   // u_i[c] + b1[c] for the block's 8 rows
  __shared__ float s_w2[NC];
  __shared__ float s_red[8][TI];   // per-wave reduction partials

  const int tid   = threadIdx.x;
  const int lane  = tid & 31;
  const int wid   = tid >> 5;
  const int i0    = blockIdx.x * TI;       // flattened row base (b*NN + n)
  const int b     = i0 >> 10;
  const int nbase = i0 & (NN - 1);
  const float* utb = ut + (size_t)b * NC * NN;

  // Stage uniform operands in LDS.
  for (int idx = tid; idx < TI * NC; idx += 256) {
    const int il = idx >> 6, c = idx & (NC - 1);
    s_vi[il][c] = utb[(size_t)c * NN + (nbase + il)] + b1[c];
  }
  if (tid < NC) s_w2[tid] = w2[tid];
  __syncthreads();

  float acc[TI][4];
#pragma unroll
  for (int il = 0; il < TI; ++il)
#pragma unroll
    for (int jj = 0; jj < 4; ++jj) acc[il][jj] = 0.0f;

  // Hot loop: per channel, 4 coalesced u_j loads feed 32 sub/max/fma chains.
#pragma unroll 4
  for (int c = 0; c < NC; ++c) {
    const float w2c = s_w2[c];
    const float* up = utb + (size_t)c * NN + tid;
    const float u0 = up[0], u1 = up[256], u2 = up[512], u3 = up[768];
#pragma unroll
    for (int il = 0; il < TI; ++il) {
      const float vic = s_vi[il][c];
      acc[il][0] += w2c * fmaxf(vic - u0, 0.0f);
      acc[il][1] += w2c * fmaxf(vic - u1, 0.0f);
      acc[il][2] += w2c * fmaxf(vic - u2, 0.0f);
      acc[il][3] += w2c * fmaxf(vic - u3, 0.0f);
    }
  }

  // Row max: shfl_xor within wave32, then LDS across the 8 waves.
#pragma unroll
  for (int il = 0; il < TI; ++il) {
    float m = fmaxf(fmaxf(acc[il][0], acc[il][1]), fmaxf(acc[il][2], acc[il][3]));
#pragma unroll
    for (int off = 16; off >= 1; off >>= 1) m = fmaxf(m, __shfl_xor(m, off, 32));
    if (lane == 0) s_red[wid][il] = m;
  }
  __syncthreads();
  float mx[TI];
#pragma unroll
  for (int il = 0; il < TI; ++il) {
    float m = s_red[0][il];
#pragma unroll
    for (int w = 1; w < 8; ++w) m = fmaxf(m, s_red[w][il]);
    mx[il] = m;
  }
  __syncthreads();  // s_red reuse

  // exp + row sum.
#pragma unroll
  for (int il = 0; il < TI; ++il) {
    float s = 0.0f;
#pragma unroll
    for (int jj = 0; jj < 4; ++jj) {
      acc[il][jj] = __expf(acc[il][jj] - mx[il]);
      s += acc[il][jj];
    }
#pragma unroll
    for (int off = 16; off >= 1; off >>= 1) s += __shfl_xor(s, off, 32);
    if (lane == 0) s_red[wid][il] = s;
  }
  __syncthreads();

  // Normalize and store (coalesced rows).
#pragma unroll
  for (int il = 0; il < TI; ++il) {
    float s = 0.0f;
#pragma unroll
    for (int w = 0; w < 8; ++w) s += s_red[w][il];
    const float inv = 1.0f / s;
    float* orow = out + (size_t)(i0 + il) * NN;
#pragma unroll
    for (int jj = 0; jj < 4; ++jj) orow[tid + jj * 256] = acc[il][jj] * inv;
  }
}

// -----------------------------------------------------------------------------
// Launch: k1 (WMMA projection, 512 waves) then k2 (pairwise + softmax).
// Workspace: ut = 2*64*1024 floats = 512 KB in d_ws.
// -----------------------------------------------------------------------------
extern "C" void kernel_launch(void* const* d_in, const int* in_sizes, int n_in,
                              void* d_out, int out_size, void* d_ws, size_t ws_size,
                              hipStream_t stream) {
  (void)in_sizes; (void)n_in; (void)out_size; (void)ws_size;
  const float* x  = (const float*)d_in[0];  // [2,1024,64]
  const float* w1 = (const float*)d_in[1];  // [64,64]
  const float* b1 = (const float*)d_in[2];  // [64]
  const float* w2 = (const float*)d_in[3];  // [64]
  // d_in[4] = b2: constant shift, cancels in softmax — unused.

  float* ut  = (float*)d_ws;   // [2][64][1024] transposed projection
  float* out = (float*)d_out;  // [2,1024,1024]

  k1_proj_wmma<<<64, 256, 0, stream>>>(x, w1, ut);          // 512 tiles / 8 waves per block
  k2_scores_softmax<<<ROWS / TI, 256, 0, stream>>>(ut, b1, w2, out);
}